// BCH_41506563948874
// MI455X (gfx1250) — compile-verified
//
#include <hip/hip_runtime.h>

// BCH(trunc=2) on a (2,3,128,128,128) f32 field with circulant bounds.
// Memory-bound 7-point stencil: ~151 MB min HBM traffic -> ~6.5 us @ 23.3 TB/s.
// Uses gfx1250 async global->LDS b128 staging (ASYNCcnt) + LDS stencil reuse.

namespace {
constexpr int N      = 128;      // X = Y = Z
constexpr int NM     = N - 1;    // wrap mask (N is a power of two)
constexpr int NLINES = 2 * 3 * 5;            // fields x channels x stencil lines = 30
constexpr int CHUNKS = NLINES * (N / 4);     // 16-byte chunks to stage = 960
}

__global__ __launch_bounds__(N)
void bch_stencil_kernel(const float* __restrict__ Lf,
                        const float* __restrict__ Rf,
                        float* __restrict__ Of) {
  // LDS layout: [field(2)][chan(3)][line(5)][z(128)] floats = 15360 bytes
  // line ids: 0 = center, 1 = x-1, 2 = x+1, 3 = y-1, 4 = y+1
  __shared__ float smem[NLINES * N];

  const int y   = blockIdx.x;
  const int x   = blockIdx.y;
  const int b   = blockIdx.z;
  const int tid = threadIdx.x;

  const int xm = (x + NM) & NM, xp = (x + 1) & NM;
  const int ym = (y + NM) & NM, yp = (y + 1) & NM;

  const int xs[5] = {x, xm, xp, x, x};
  const int ys[5] = {y, y,  y,  ym, yp};

  // ---- Stage 30 z-lines into LDS with async global->LDS b128 transfers ----
  // 960 chunks over 128 threads: 7 full passes + one pass for waves 0-1 only,
  // so every issued async instruction runs with a full wave32 EXEC group.
  for (int c = tid; c < CHUNKS; c += N) {
    const int line_id = c >> 5;          // 0..29
    const int lane    = c & 31;          // which 16B chunk of the 512B line
    const int f   = line_id / 15;        // 0 = left field, 1 = right field
    const int rem = line_id - f * 15;
    const int d   = rem / 5;             // channel
    const int ln  = rem - d * 5;         // stencil line

    const float* base = f ? Rf : Lf;
    const size_t row  = ((((size_t)(b * 3 + d) * N) + xs[ln]) * N + ys[ln]) * N;
    const float* src  = base + row + (size_t)lane * 4;

    // Low 32 bits of a flat pointer to __shared__ == wave-relative LDS byte addr.
    unsigned lds_addr = (unsigned)(uintptr_t)(&smem[line_id * N + lane * 4]);
    unsigned long long gaddr = (unsigned long long)(uintptr_t)src;

    asm volatile("global_load_async_to_lds_b128 %0, %1, off"
                 :
                 : "v"(lds_addr), "v"(gaddr)
                 : "memory");
  }
  asm volatile("s_wait_asynccnt 0" ::: "memory");  // my wave's LDS writes landed
  __syncthreads();                                 // everyone's landed

  // ---- Compute: each thread owns one z of the pencil, all 3 channels ----
  const int z  = tid;
  const int zp = (z + 1) & NM;
  const int zm = (z + NM) & NM;

  auto SM = [&](int f, int d, int ln, int zz) -> float {
    return smem[((f * 3 + d) * 5 + ln) * N + zz];
  };

  float lC[3], rC[3];
#pragma unroll
  for (int d = 0; d < 3; ++d) {
    lC[d] = SM(0, d, 0, z);
    rC[d] = SM(1, d, 0, z);
  }

#pragma unroll
  for (int i = 0; i < 3; ++i) {
    // j = 0 (x-derivative), j = 1 (y), j = 2 (z, from shifted center line)
    float acc;
    acc  = (SM(0,i,2,z)  - SM(0,i,1,z))  * rC[0] - (SM(1,i,2,z)  - SM(1,i,1,z))  * lC[0];
    acc += (SM(0,i,4,z)  - SM(0,i,3,z))  * rC[1] - (SM(1,i,4,z)  - SM(1,i,3,z))  * lC[1];
    acc += (SM(0,i,0,zp) - SM(0,i,0,zm)) * rC[2] - (SM(1,i,0,zp) - SM(1,i,0,zm)) * lC[2];

    const size_t o = ((((size_t)(b * 3 + i) * N) + x) * N + y) * N + z;
    Of[o] = lC[i] + rC[i] + 0.25f * acc;   // l + r + 0.5*(0.5*central diff) bracket
  }
}

extern "C" void kernel_launch(void* const* d_in, const int* in_sizes, int n_in,
                              void* d_out, int out_size, void* d_ws, size_t ws_size,
                              hipStream_t stream) {
  const float* L = (const float*)d_in[0];   // left  (2,3,128,128,128) f32
  const float* R = (const float*)d_in[1];   // right (2,3,128,128,128) f32
  float* O = (float*)d_out;                 // (2,3,128,128,128) f32

  dim3 grid(N, N, 2);   // (y, x, b)
  dim3 block(N);        // one thread per z; 4 wave32s
  bch_stencil_kernel<<<grid, block, 0, stream>>>(L, R, O);
}